// PruningAwareContrastiveLearning_7352984011020
// MI455X (gfx1250) — compile-verified
//
#include <hip/hip_runtime.h>
#include <hip/hip_bf16.h>
#include <math.h>

// ---------------- constants from the reference ----------------
#define TAU    0.4f
#define ALPHA  1.0f
#define GAMMA_ 0.7f
#define DELTA_ 0.6f
#define MU_    0.6f
#define NU_    0.3f
#define N_     8192
#define E_     2048
#define D_     256
#define SAMP_  100
#define M_FN_  100
#define M_HN_  200
#define ECAP_  500

typedef float v2f __attribute__((ext_vector_type(2)));
typedef float v8f __attribute__((ext_vector_type(8)));

#if defined(__gfx1250__) && __has_builtin(__builtin_amdgcn_wmma_f32_16x16x4_f32)
#define USE_WMMA_F32 1
#else
#define USE_WMMA_F32 0
#endif

// ---------------- block helpers (blockDim.x == 256) ----------------
__device__ __forceinline__ float block_sum256(float v, float* sbuf) {
  int t = threadIdx.x;
  sbuf[t] = v; __syncthreads();
#pragma unroll
  for (int o = 128; o > 0; o >>= 1) {
    if (t < o) sbuf[t] += sbuf[t + o];
    __syncthreads();
  }
  float r = sbuf[0];
  __syncthreads();
  return r;
}

__device__ __forceinline__ float block_max256(float v, float* sbuf) {
  int t = threadIdx.x;
  sbuf[t] = v; __syncthreads();
#pragma unroll
  for (int o = 128; o > 0; o >>= 1) {
    if (t < o) sbuf[t] = fmaxf(sbuf[t], sbuf[t + o]);
    __syncthreads();
  }
  float r = sbuf[0];
  __syncthreads();
  return r;
}

// online logsumexp update: state (m, s) means  sum(exp(x - m)) = s
__device__ __forceinline__ void olse(float& m, float& s, float v) {
  if (v > m) { s = s * expf(m - v) + 1.f; m = v; }
  else       { s += expf(v - m); }
}

// 256-length dot product with b128 loads (a is 16B-aligned LDS, b is global)
__device__ __forceinline__ float dot256(const float* __restrict__ a,
                                        const float* __restrict__ b) {
  const float4* a4 = (const float4*)a;
  const float4* b4 = (const float4*)b;
  float d = 0.f;
#pragma unroll 8
  for (int k = 0; k < D_ / 4; ++k) {
    float4 x = a4[k], y = b4[k];
    d = fmaf(x.x, y.x, d); d = fmaf(x.y, y.y, d);
    d = fmaf(x.z, y.z, d); d = fmaf(x.w, y.w, d);
  }
  return d;
}

// scalar accumulator layout in ws
// [0]=nz [1]=nzp [2]=dsq [3]=osq [4]=cl_sum [5]=fn_sum [6]=n_fn [7]=hn_sum [8]=n_hn

// ---------------- 0: zero scalars ----------------
__global__ void init_kernel(float* scal) {
  if (threadIdx.x < 16) scal[threadIdx.x] = 0.f;
}

// ---------------- 1: nonzero counts (b128 loads, HBM-rate) ----------------
__global__ void nnz_kernel(const float4* __restrict__ Ho4, const float4* __restrict__ Hp4,
                           float* scal) {
  __shared__ float sbuf[256];
  const size_t total4 = (size_t)N_ * (size_t)E_ / 4;
  float co = 0.f, cp = 0.f;
  for (size_t idx = (size_t)blockIdx.x * blockDim.x + threadIdx.x; idx < total4;
       idx += (size_t)gridDim.x * blockDim.x) {
    float4 o = Ho4[idx];
    float4 p = Hp4[idx];
    co += ((o.x != 0.f) ? 1.f : 0.f) + ((o.y != 0.f) ? 1.f : 0.f) +
          ((o.z != 0.f) ? 1.f : 0.f) + ((o.w != 0.f) ? 1.f : 0.f);
    cp += ((p.x != 0.f) ? 1.f : 0.f) + ((p.y != 0.f) ? 1.f : 0.f) +
          ((p.z != 0.f) ? 1.f : 0.f) + ((p.w != 0.f) ? 1.f : 0.f);
  }
  float so = block_sum256(co, sbuf);
  float sp = block_sum256(cp, sbuf);
  if (threadIdx.x == 0) { atomicAdd(&scal[0], so); atomicAdd(&scal[1], sp); }
}

// ---------------- 2: topo term  ||A_s - Ap_s||^2 and ||A_s||^2 ----------------
__global__ void topo_kernel(const float* __restrict__ Ho, const float* __restrict__ Hp,
                            float* scal) {
  __shared__ float sbuf[256];
  float d2 = 0.f, o2 = 0.f;
  for (int p = blockIdx.x * 256 + threadIdx.x; p < SAMP_ * SAMP_; p += gridDim.x * 256) {
    int i = p / SAMP_, j = p % SAMP_;
    const float4* hi = (const float4*)(Ho + (size_t)i * E_);
    const float4* hj = (const float4*)(Ho + (size_t)j * E_);
    const float4* pi = (const float4*)(Hp + (size_t)i * E_);
    const float4* pj = (const float4*)(Hp + (size_t)j * E_);
    float a = 0.f, ap = 0.f;
#pragma unroll 5
    for (int k = 0; k < SAMP_ / 4; ++k) {
      float4 x = hi[k], y = hj[k];
      a = fmaf(x.x, y.x, a); a = fmaf(x.y, y.y, a);
      a = fmaf(x.z, y.z, a); a = fmaf(x.w, y.w, a);
      float4 u = pi[k], v = pj[k];
      ap = fmaf(u.x, v.x, ap); ap = fmaf(u.y, v.y, ap);
      ap = fmaf(u.z, v.z, ap); ap = fmaf(u.w, v.w, ap);
    }
    float df = a - ap;
    d2 = fmaf(df, df, d2);
    o2 = fmaf(a, a, o2);
  }
  d2 = block_sum256(d2, sbuf);
  o2 = block_sum256(o2, sbuf);
  if (threadIdx.x == 0) { atomicAdd(&scal[2], d2); atomicAdd(&scal[3], o2); }
}

// ---------------- 3: A100 / Ap100 (full K=2048, b128 loads) ----------------
__global__ void a100_kernel(const float* __restrict__ Ho, const float* __restrict__ Hp,
                            float* __restrict__ A100, float* __restrict__ Ap100) {
  int p = blockIdx.x * 256 + threadIdx.x;
  if (p >= M_FN_ * M_FN_) return;
  int i = p / M_FN_, j = p % M_FN_;
  const float4* hi = (const float4*)(Ho + (size_t)i * E_);
  const float4* hj = (const float4*)(Ho + (size_t)j * E_);
  const float4* pi = (const float4*)(Hp + (size_t)i * E_);
  const float4* pj = (const float4*)(Hp + (size_t)j * E_);
  float a = 0.f, ap = 0.f;
#pragma unroll 4
  for (int k = 0; k < E_ / 4; ++k) {
    float4 x = hi[k], y = hj[k];
    a = fmaf(x.x, y.x, a); a = fmaf(x.y, y.y, a);
    a = fmaf(x.z, y.z, a); a = fmaf(x.w, y.w, a);
    float4 u = pi[k], v = pj[k];
    ap = fmaf(u.x, v.x, ap); ap = fmaf(u.y, v.y, ap);
    ap = fmaf(u.z, v.z, ap); ap = fmaf(u.w, v.w, ap);
  }
  A100[p] = a;
  Ap100[p] = ap;
}

// ---------------- 4: fused 2-layer MLP, 16 rows per wave, WMMA f32 16x16x4 ----------------
__global__ void __launch_bounds__(32)
mlp16_kernel(const float* __restrict__ X,
             const float* __restrict__ W1, const float* __restrict__ b1,
             const float* __restrict__ W2, const float* __restrict__ b2,
             float* __restrict__ Y) {
  __shared__ __align__(16) float hidden[16 * D_];
  const int row0 = blockIdx.x * 16;
#if USE_WMMA_F32
  const int lane = threadIdx.x;
  const int half = lane >> 4;      // 0 or 1
  const int l15  = lane & 15;
  const float* A = X + (size_t)row0 * D_;
  // layer 1: hidden = relu(X @ W1 + b1)
  for (int nt = 0; nt < D_; nt += 16) {
    v8f acc = {};
    for (int k = 0; k < D_; k += 4) {
      v2f a, b;
      // A 16x4 frag: lane l15+16*half holds A[l15][k+2*half + {0,1}]
      a.x = A[(size_t)l15 * D_ + k + 2 * half + 0];
      a.y = A[(size_t)l15 * D_ + k + 2 * half + 1];
      // B 4x16 frag: lane n+16*(kk>>1) holds B[kk][n], vgpr = kk&1
      b.x = W1[(size_t)(k + 2 * half + 0) * D_ + nt + l15];
      b.y = W1[(size_t)(k + 2 * half + 1) * D_ + nt + l15];
      acc = __builtin_amdgcn_wmma_f32_16x16x4_f32(false, a, false, b, (short)0, acc,
                                                  false, false);
    }
#pragma unroll
    for (int r = 0; r < 8; ++r)
      hidden[(r + 8 * half) * D_ + nt + l15] = fmaxf(acc[r] + b1[nt + l15], 0.f);
  }
  __syncthreads();
  // layer 2: Y = hidden @ W2 + b2
  for (int nt = 0; nt < D_; nt += 16) {
    v8f acc = {};
    for (int k = 0; k < D_; k += 4) {
      v2f a, b;
      a.x = hidden[l15 * D_ + k + 2 * half + 0];
      a.y = hidden[l15 * D_ + k + 2 * half + 1];
      b.x = W2[(size_t)(k + 2 * half + 0) * D_ + nt + l15];
      b.y = W2[(size_t)(k + 2 * half + 1) * D_ + nt + l15];
      acc = __builtin_amdgcn_wmma_f32_16x16x4_f32(false, a, false, b, (short)0, acc,
                                                  false, false);
    }
#pragma unroll
    for (int r = 0; r < 8; ++r)
      Y[(size_t)(row0 + r + 8 * half) * D_ + nt + l15] = acc[r] + b2[nt + l15];
  }
#else
  // scalar fallback (only if the f32 WMMA builtin is unavailable)
  const int lane = threadIdx.x;
  for (int c = lane; c < D_; c += 32)
    for (int r = 0; r < 16; ++r) {
      float s = b1[c];
      for (int k = 0; k < D_; ++k) s = fmaf(X[(size_t)(row0 + r) * D_ + k], W1[(size_t)k * D_ + c], s);
      hidden[r * D_ + c] = fmaxf(s, 0.f);
    }
  __syncthreads();
  for (int c = lane; c < D_; c += 32)
    for (int r = 0; r < 16; ++r) {
      float s = b2[c];
      for (int k = 0; k < D_; ++k) s = fmaf(hidden[r * D_ + k], W2[(size_t)k * D_ + c], s);
      Y[(size_t)(row0 + r) * D_ + c] = s;
    }
#endif
}

// ---------------- 5: per-row inverse L2 norm (l2n eps 1e-8) ----------------
__global__ void invnorm_kernel(const float* __restrict__ X, float* __restrict__ inv) {
  __shared__ float sbuf[256];
  int r = blockIdx.x, t = threadIdx.x;
  float v = X[(size_t)r * D_ + t];
  float s = block_sum256(v * v, sbuf);
  if (t == 0) inv[r] = 1.f / fmaxf(sqrtf(s), 1e-8f);
}

// ---------------- 6: edge pooling  edge_mean = mask.T @ z / max(cnt,1) ----------------
// Stage 256-row column tiles of H_pruned in LDS (parallel strided loads, L2-resident),
// then scan the tile with b128 LDS broadcasts.
__global__ void edge_kernel(const float* __restrict__ Hp, const float* __restrict__ z,
                            float* __restrict__ emean) {
  __shared__ __align__(16) float hcol[256];
  int e = blockIdx.x, t = threadIdx.x;
  float acc = 0.f, c = 0.f;
  if (e < ECAP_) {
    for (int n0 = 0; n0 < N_; n0 += 256) {
      __syncthreads();
      hcol[t] = Hp[(size_t)(n0 + t) * E_ + e];   // 256 parallel loads
      __syncthreads();
      const float4* hc4 = (const float4*)hcol;
#pragma unroll 4
      for (int q = 0; q < 64; ++q) {
        float4 h4 = hc4[q];                       // LDS broadcast
        int nb = n0 + 4 * q;
        if (h4.x > 0.f) { acc += z[(size_t)(nb + 0) * D_ + t]; c += 1.f; }
        if (h4.y > 0.f) { acc += z[(size_t)(nb + 1) * D_ + t]; c += 1.f; }
        if (h4.z > 0.f) { acc += z[(size_t)(nb + 2) * D_ + t]; c += 1.f; }
        if (h4.w > 0.f) { acc += z[(size_t)(nb + 3) * D_ + t]; c += 1.f; }
      }
    }
  }
  emean[(size_t)e * D_ + t] = (c > 0.f) ? acc / fmaxf(c, 1.f) : 0.f;
}

// ---------------- 7: cl_loss  (row-wise log-softmax diag of 500x500 logits) ----------------
__global__ void cl_kernel(const float* __restrict__ z, const float* __restrict__ zinv,
                          const float* __restrict__ h, const float* __restrict__ hinv,
                          float* scal) {
  __shared__ __align__(16) float zi[256];
  __shared__ float sbuf[256];
  int i = blockIdx.x, t = threadIdx.x;
  zi[t] = z[(size_t)i * D_ + t];
  __syncthreads();
  float invi = zinv[i];
  float l0 = -1e30f, l1 = -1e30f;
  l0 = dot256(zi, h + (size_t)t * D_) * invi * hinv[t] / TAU;
  if (t + 256 < ECAP_)
    l1 = dot256(zi, h + (size_t)(t + 256) * D_) * invi * hinv[t + 256] / TAU;
  float M = block_max256(fmaxf(l0, l1), sbuf);
  float S = block_sum256(expf(l0 - M) + expf(l1 - M), sbuf);
  float lse = M + logf(S);
  if (t == i)        atomicAdd(&scal[4], l0 - lse);
  if (t + 256 == i)  atomicAdd(&scal[4], l1 - lse);
}

// ------- 8: fused att stats (softmax over 8192), att[:, :200], sim, lse_all -------
__global__ void att_sim_kernel(const float* __restrict__ z, const float* __restrict__ zinv,
                               float* __restrict__ sim, float* __restrict__ att,
                               float* __restrict__ lseall) {
  __shared__ __align__(16) float zi[256];
  __shared__ float rm[256];
  __shared__ float rs[256];
  __shared__ float bM, bS;
  int i = blockIdx.x, t = threadIdx.x;
  zi[t] = z[(size_t)i * D_ + t];
  __syncthreads();
  float invi = zinv[i];
  float m1 = -1e30f, s1 = 0.f;   // raw scores z_i . z_j   (for att softmax)
  float m2 = -1e30f, s2 = 0.f;   // normalized zn_i . zn_j / TAU (for lse_all)
  for (int j = t; j < N_; j += 256) {
    float d = dot256(zi, z + (size_t)j * D_);
    olse(m1, s1, d);
    olse(m2, s2, d * invi * zinv[j] / TAU);
  }
  // reduce (m1,s1)
  rm[t] = m1; rs[t] = s1; __syncthreads();
  for (int o = 128; o > 0; o >>= 1) {
    if (t < o) {
      float M = fmaxf(rm[t], rm[t + o]);
      rs[t] = rs[t] * expf(rm[t] - M) + rs[t + o] * expf(rm[t + o] - M);
      rm[t] = M;
    }
    __syncthreads();
  }
  if (t == 0) { bM = rm[0]; bS = rs[0]; }
  __syncthreads();
  float Mr = bM, Sr = bS;
  // reduce (m2,s2)
  rm[t] = m2; rs[t] = s2; __syncthreads();
  for (int o = 128; o > 0; o >>= 1) {
    if (t < o) {
      float M = fmaxf(rm[t], rm[t + o]);
      rs[t] = rs[t] * expf(rm[t] - M) + rs[t + o] * expf(rm[t + o] - M);
      rm[t] = M;
    }
    __syncthreads();
  }
  if (t == 0 && i < M_FN_) lseall[i] = rm[0] + logf(rs[0]);
  __syncthreads();
  // second pass: materialize att[:, :200] and sim (200x200)
  if (t < M_HN_) {
    float d = dot256(zi, z + (size_t)t * D_);
    att[i * M_HN_ + t] = expf(d - Mr) / Sr;
    sim[i * M_HN_ + t] = d * invi * zinv[t];
  }
}

// ---------------- 9: false-negative loss ----------------
__global__ void fn_kernel(const float* __restrict__ sim, const float* __restrict__ att,
                          const float* __restrict__ A100, const float* __restrict__ Ap100,
                          const float* __restrict__ lseall, float* scal) {
  __shared__ float sbuf[256];
  float s = 0.f, c = 0.f;
  for (int p = blockIdx.x * 256 + threadIdx.x; p < M_FN_ * M_FN_; p += gridDim.x * 256) {
    int i = p / M_FN_, j = p % M_FN_;
    if (j > i) {
      float sv = sim[i * M_HN_ + j];
      float av = att[i * M_HN_ + j];
      bool m = (Ap100[p] == 0.f) && (A100[p] > 0.f) && (sv > GAMMA_) && (av > DELTA_);
      if (m) { s += -(sv / TAU - lseall[i]); c += 1.f; }
    }
  }
  s = block_sum256(s, sbuf);
  c = block_sum256(c, sbuf);
  if (threadIdx.x == 0) { atomicAdd(&scal[5], s); atomicAdd(&scal[6], c); }
}

// ---------------- 10: hard-negative loss ----------------
__global__ void hard_kernel(const float* __restrict__ sim, const float* __restrict__ att,
                            float* scal) {
  __shared__ float sbuf[256];
  int i = blockIdx.x, t = threadIdx.x;
  float sv = 0.f;
  bool mask = false;
  float v = -1e30f;                 // matches reference's where(mask, neg, -1e30)
  if (t < M_HN_) {
    sv = sim[i * M_HN_ + t];
    float av = att[i * M_HN_ + t];
    mask = (t != i) && (sv > MU_) && (av < NU_);
    if (mask) v = -sv / TAU;
  }
  float M = block_max256(v, sbuf);
  float S = block_sum256(expf(v - M), sbuf);
  float lse = M + logf(S);
  float contrib = 0.f, c = 0.f;
  if (mask) {
    float w = fminf(sv / MU_, 1.f);
    contrib = -w * (v - lse);
    c = 1.f;
  }
  contrib = block_sum256(contrib, sbuf);
  c = block_sum256(c, sbuf);
  if (t == 0) { atomicAdd(&scal[7], contrib); atomicAdd(&scal[8], c); }
}

// ---------------- 11: finalize scalar ----------------
__global__ void finalize_kernel(const float* scal, const float* __restrict__ bw,
                                float* __restrict__ out) {
  float nz = scal[0], nzp = scal[1], dsq = scal[2], osq = scal[3];
  float sparsity = (nz > 0.f) ? nzp / fmaxf(nz, 1.f) : 1.f;
  float dn = sqrtf(dsq), on = sqrtf(osq);
  float topo = (on > 0.f) ? expf(-ALPHA * dn / fmaxf(on, 1e-12f)) : 1.f;
  float retention = sparsity * topo;
  float cl = -scal[4] / (float)ECAP_;
  float n_fn = scal[6];
  float fnl = (n_fn > 0.f) ? scal[5] / fmaxf(n_fn, 1.f) : 0.f;
  float n_hn = scal[8];
  float hl = (n_hn > 0.f) ? scal[7] / fmaxf(n_hn, 1.f) : 0.f;
  out[0] = bw[0] * (retention * cl + fnl + hl);
}

// ---------------- launcher ----------------
extern "C" void kernel_launch(void* const* d_in, const int* in_sizes, int n_in,
                              void* d_out, int out_size, void* d_ws, size_t ws_size,
                              hipStream_t stream) {
  (void)in_sizes; (void)n_in; (void)out_size; (void)ws_size;
  const float* Ho  = (const float*)d_in[0];   // H_original  (8192, 2048)
  const float* Hp  = (const float*)d_in[1];   // H_pruned    (8192, 2048)
  const float* NE  = (const float*)d_in[2];   // node_embeddings (8192, 256)
  const float* BW  = (const float*)d_in[3];   // behavior_weights (1,)
  const float* Wn1 = (const float*)d_in[4];
  const float* bn1 = (const float*)d_in[5];
  const float* Wn2 = (const float*)d_in[6];
  const float* bn2 = (const float*)d_in[7];
  const float* We1 = (const float*)d_in[8];
  const float* be1 = (const float*)d_in[9];
  const float* We2 = (const float*)d_in[10];
  const float* be2 = (const float*)d_in[11];

  float* ws     = (float*)d_ws;
  float* z      = ws;                       // 8192*256
  float* zinv   = z      + (size_t)N_ * D_; // 8192
  float* emean  = zinv   + N_;              // 512*256
  float* hedg   = emean  + 512 * D_;        // 512*256
  float* hinv   = hedg   + 512 * D_;        // 512
  float* sim    = hinv   + 512;             // 200*200
  float* att    = sim    + M_HN_ * M_HN_;   // 200*200
  float* lseall = att    + M_HN_ * M_HN_;   // 128
  float* A100   = lseall + 128;             // 100*100
  float* Ap100  = A100   + M_FN_ * M_FN_;   // 100*100
  float* scal   = Ap100  + M_FN_ * M_FN_;   // 16

  init_kernel<<<1, 32, 0, stream>>>(scal);
  nnz_kernel<<<2048, 256, 0, stream>>>((const float4*)Ho, (const float4*)Hp, scal);
  topo_kernel<<<40, 256, 0, stream>>>(Ho, Hp, scal);
  a100_kernel<<<40, 256, 0, stream>>>(Ho, Hp, A100, Ap100);

  mlp16_kernel<<<N_ / 16, 32, 0, stream>>>(NE, Wn1, bn1, Wn2, bn2, z);        // z
  invnorm_kernel<<<N_, 256, 0, stream>>>(z, zinv);

  edge_kernel<<<512, 256, 0, stream>>>(Hp, z, emean);
  mlp16_kernel<<<512 / 16, 32, 0, stream>>>(emean, We1, be1, We2, be2, hedg); // h_edges
  invnorm_kernel<<<512, 256, 0, stream>>>(hedg, hinv);

  cl_kernel<<<ECAP_, 256, 0, stream>>>(z, zinv, hedg, hinv, scal);
  att_sim_kernel<<<M_HN_, 256, 0, stream>>>(z, zinv, sim, att, lseall);
  fn_kernel<<<40, 256, 0, stream>>>(sim, att, A100, Ap100, lseall, scal);
  hard_kernel<<<M_HN_, 256, 0, stream>>>(sim, att, scal);

  finalize_kernel<<<1, 1, 0, stream>>>(scal, BW, (float*)d_out);
}